// LagAwareWaveformLoss_58093727645997
// MI455X (gfx1250) — compile-verified
//
#include <hip/hip_runtime.h>

// LagAwareWaveformLoss for MI455X (gfx1250, wave32).
// B=512 samples, C=1, L=4096, lags in [-20,20] (41), tau=0.05.
// Strategy:
//   - Stage pred[b,:], gt[b,:] in LDS (32KB).
//   - Cross-correlation via fp32 WMMA 16x16x4: lag s = 16 + m - 16n.
//     j in [0,4044] step 4 (1012 K-steps), contiguous chunk per wave,
//     unrolled 4x with 4 independent accumulators (no WMMA->WMMA RAW chain,
//     DS loads pipelined). Per-lag head/tail corrections (<48 elems) on VALU
//     make each lag's window match the reference overlap exactly.
//   - Per-lag sums (Sp,Sp2,Sg,Sg2,S|p-g|) via wave-strided loops + shfl reduce.
//   - Thread 0: argmax / softmax combine -> 4 values per sample into d_ws.
//   - Second kernel: deterministic tree reduction over 512 samples -> d_out[4].

#define BATCH  512
#define LEN    4096
#define NLAG   41
#define MAXLAG 20
#define TAU    0.05f
#define NWAVE  8
#define NSTEP  1012   // j = 4*t, t in [0,1012) -> j in [0,4044]
#define CHUNK  127    // ceil(1012/8); wave 7 gets the 123-step remainder

typedef __attribute__((ext_vector_type(2))) float v2f;
typedef __attribute__((ext_vector_type(8))) float v8f;

__device__ __forceinline__ float waveReduce(float v) {
#pragma unroll
    for (int o = 16; o > 0; o >>= 1) v += __shfl_xor(v, o, 32);
    return v;
}

__global__ __launch_bounds__(256)
void lagloss_main(const float* __restrict__ pred,
                  const float* __restrict__ gt,
                  float* __restrict__ ws) {
    __shared__ float predS[LEN];
    __shared__ float gtS[LEN];
    __shared__ float crossW[NWAVE][16][4];
    __shared__ float lagCross[NLAG];
    __shared__ float lagPcc[NLAG];
    __shared__ float lagL1[NLAG];

    const int b    = blockIdx.x;
    const int tid  = threadIdx.x;
    const int lane = tid & 31;
    const int wave = tid >> 5;
    const int mrow = lane & 15;   // M row (A) / N col (B,C,D) for this lane
    const int hi   = lane >> 4;   // lane group: selects K pair

    // ---- Phase 0: stage sample into LDS ----
    const float* pb = pred + (size_t)b * LEN;
    const float* gb = gt   + (size_t)b * LEN;
    for (int i = tid; i < LEN; i += 256) { predS[i] = pb[i]; gtS[i] = gb[i]; }
    __syncthreads();

    // ---- Phase 1: fp32 WMMA cross-correlation ----
    // A[m,k] = pred[j+16+k+m], B[k,n] = gt[j+k+16n]
    // D[m,n] accumulates sum over i in [16n, 4048+16n) of pred[i+s]*gt[i],
    // with lag s = 16 + m - 16n. Only n=0..3 columns are meaningful
    // (higher-n B reads are clamped in-bounds and ignored).
    auto wstep = [&](int t, v8f c) -> v8f {
        const int j  = t << 2;
        const int ab = j + 16 + 2 * hi + mrow;         // max 4077 (+1 -> 4078)
        v2f a;
        a.x = predS[ab];
        a.y = predS[ab + 1];
        int bb  = j + 2 * hi + (mrow << 4);            // up to 4286 for n=15
        int bi0 = bb     < (LEN - 1) ? bb     : (LEN - 1);
        int bi1 = bb + 1 < (LEN - 1) ? bb + 1 : (LEN - 1);
        v2f bv;
        bv.x = gtS[bi0];
        bv.y = gtS[bi1];
        return __builtin_amdgcn_wmma_f32_16x16x4_f32(
            /*neg_a=*/false, a, /*neg_b=*/false, bv,
            /*c_mod=*/(short)0, c, /*reuse_a=*/false, /*reuse_b=*/false);
    };

    v8f acc0 = {}, acc1 = {}, acc2 = {}, acc3 = {};
    const int tbeg = wave * CHUNK;
    int tend = tbeg + CHUNK;
    if (tend > NSTEP) tend = NSTEP;
    int t = tbeg;
    for (; t + 4 <= tend; t += 4) {
        acc0 = wstep(t + 0, acc0);   // independent accumulators:
        acc1 = wstep(t + 1, acc1);   // WMMAs issue back-to-back,
        acc2 = wstep(t + 2, acc2);   // DS loads stay pipelined
        acc3 = wstep(t + 3, acc3);
    }
    for (; t < tend; ++t) acc0 = wstep(t, acc0);

    v8f acc;
#pragma unroll
    for (int r = 0; r < 8; ++r) acc[r] = (acc0[r] + acc1[r]) + (acc2[r] + acc3[r]);

    // D layout: VGPR r, lanes 0-15: row M=r, col N=lane; lanes 16-31: M=r+8.
    if (mrow < 4) {
#pragma unroll
        for (int r = 0; r < 8; ++r)
            crossW[wave][r + 8 * hi][mrow] = acc[r];
    }
    __syncthreads();
    if (tid < 64) {
        const int m = tid >> 2, col = tid & 3;
        float s = 0.f;
#pragma unroll
        for (int w = 0; w < NWAVE; ++w) s += crossW[w][m][col];
        const int lag = 16 + m - (col << 4);
        if (lag >= -MAXLAG && lag <= MAXLAG) lagCross[lag + MAXLAG] = s;
    }
    __syncthreads();

    // ---- Phase 2: per-lag sums + WMMA window corrections ----
    for (int li = wave; li < NLAG; li += NWAVE) {
        const int s  = li - MAXLAG;
        const int i0 = (s < 0) ? -s : 0;
        const int i1 = LEN - ((s > 0) ? s : 0);
        const int n  = (s >= 16) ? 0 : (s >= 0) ? 1 : (s >= -16) ? 2 : 3;
        const int c0 = n << 4;            // WMMA-covered interval [c0, c1)
        const int c1 = 4048 + (n << 4);

        float sp = 0.f, sp2 = 0.f, sg = 0.f, sg2 = 0.f, sl1 = 0.f, ce = 0.f;
        for (int i = i0 + lane; i < i1; i += 32) {
            const float p = predS[i + s];
            const float g = gtS[i];
            sp  += p;     sp2 += p * p;
            sg  += g;     sg2 += g * g;
            sl1 += fabsf(p - g);
        }
        // head correction: [i0, c0)
        const int h1 = (c0 < i1) ? c0 : i1;
        for (int i = i0 + lane; i < h1; i += 32) ce += predS[i + s] * gtS[i];
        // tail correction: [c1, i1)
        const int t0 = (c1 > i0) ? c1 : i0;
        for (int i = t0 + lane; i < i1; i += 32) ce += predS[i + s] * gtS[i];

        sp  = waveReduce(sp);   sp2 = waveReduce(sp2);
        sg  = waveReduce(sg);   sg2 = waveReduce(sg2);
        sl1 = waveReduce(sl1);  ce  = waveReduce(ce);

        if (lane == 0) {
            const float Ln   = (float)(i1 - i0);
            const float num  = lagCross[li] + ce - sp * sg / Ln;
            const float vp   = sp2 - sp * sp / Ln;
            const float vg   = sg2 - sg * sg / Ln;
            float den2 = vp * vg;
            if (den2 < 1e-12f) den2 = 1e-12f;
            lagPcc[li] = num / sqrtf(den2);
            lagL1[li]  = sl1 / Ln;
        }
    }
    __syncthreads();

    // ---- Phase 3: argmax + softmax combine (one thread, 41 entries) ----
    if (tid == 0) {
        float bestP = -1e30f;
        int   bestI = 0;
        for (int li = 0; li < NLAG; ++li) {
            const float p = lagPcc[li];
            if (p > bestP) { bestP = p; bestI = li; }
        }
        float wsum = 0.f, pen = 0.f;
        for (int li = 0; li < NLAG; ++li) {
            const float e = __expf((lagPcc[li] - bestP) * (1.0f / TAU));
            wsum += e;
            pen  += e * (fabsf((float)(li - MAXLAG)) * (1.0f / (float)MAXLAG));
        }
        float* o = ws + (size_t)b * 4;
        o[0] = 1.0f - bestP;
        o[1] = lagL1[bestI];
        o[2] = 1.0f - lagPcc[MAXLAG];
        o[3] = pen / wsum;
    }
}

__global__ __launch_bounds__(256)
void lagloss_final(const float* __restrict__ ws, float* __restrict__ out) {
    __shared__ float red[256][4];
    const int t = threadIdx.x;
    float s0 = 0.f, s1 = 0.f, s2 = 0.f, s3 = 0.f;
    for (int b = t; b < BATCH; b += 256) {
        s0 += ws[b * 4 + 0];
        s1 += ws[b * 4 + 1];
        s2 += ws[b * 4 + 2];
        s3 += ws[b * 4 + 3];
    }
    red[t][0] = s0; red[t][1] = s1; red[t][2] = s2; red[t][3] = s3;
    __syncthreads();
    for (int stride = 128; stride > 0; stride >>= 1) {
        if (t < stride) {
#pragma unroll
            for (int k = 0; k < 4; ++k) red[t][k] += red[t + stride][k];
        }
        __syncthreads();
    }
    if (t < 4) out[t] = red[0][t] * (1.0f / (float)BATCH);
}

extern "C" void kernel_launch(void* const* d_in, const int* in_sizes, int n_in,
                              void* d_out, int out_size, void* d_ws, size_t ws_size,
                              hipStream_t stream) {
    const float* pred = (const float*)d_in[0];
    const float* gt   = (const float*)d_in[1];
    float* out = (float*)d_out;
    float* wsf = (float*)d_ws;   // BATCH*4 floats = 8 KB scratch

    lagloss_main<<<BATCH, 256, 0, stream>>>(pred, gt, wsf);
    lagloss_final<<<1, 256, 0, stream>>>(wsf, out);
}